// Model_74440373174851
// MI455X (gfx1250) — compile-verified
//
#include <hip/hip_runtime.h>

#ifndef __has_builtin
#define __has_builtin(x) 0
#endif

// ---------------------------------------------------------------------------
// 20-step periodic 2D Jacobi on 8192x8192 fp32, MI455X (gfx1250, wave32).
//
// Memory-bound (AI ~0.6 FLOP/byte) -> temporal blocking: 4 launches x 5 fused
// steps in LDS. Per workgroup: async-load (b128, ASYNCcnt) a 138-row x 144-col
// halo tile (center 128x128 at rows 5..132, cols 8..135; left halo aligned to
// a global 4-float boundary so every b128 transfer is 16B-aligned and never
// straddles the periodic seam, since 4 | 8192), run 5 ping-pong sweeps with
// packed-f32 math (v_pk_add_f32 via ext_vector float2, 2 rows per lane),
// store the center with float4. HBM traffic: 10.2 GB naive -> ~2.3 GB.
// LDS: 2 x 79,488 B = 158,976 B -> 2 workgroups per 320 KB WGP.
// ---------------------------------------------------------------------------

namespace {
constexpr int    N        = 8192;
constexpr int    MASK     = N - 1;            // power-of-two periodic wrap
constexpr int    T        = 5;                // fused steps per launch (20 = 4*5)
constexpr int    BX       = 128;              // output tile edge (divides 8192)
constexpr int    TH       = BX + 2 * T;       // 138 rows (5-halo top/bottom)
constexpr int    TX       = 144;              // LDS row stride: 8+128+8 cols
constexpr int    CGRP     = TX / 4;           // 36 b128 column groups per row
constexpr int    NTILES   = N / BX;           // 64
constexpr int    THREADS  = 256;              // 8 wave32 waves
constexpr size_t SMEM_BYTES = size_t(2) * TX * TH * sizeof(float); // 158,976 B
}

typedef int v4i __attribute__((vector_size(16)));        // matches builtin sig
typedef __attribute__((address_space(1))) v4i glob_v4i;
typedef __attribute__((address_space(3))) v4i lds_v4i;
typedef float v2f __attribute__((ext_vector_type(2)));

// ---- CDNA5 async global->LDS b128 load (16 B per lane, ASYNCcnt path) -----
__device__ __forceinline__ void async_load_b128_to_lds(const float* gp, float* lp) {
#if __has_builtin(__builtin_amdgcn_global_load_async_to_lds_b128)
  __builtin_amdgcn_global_load_async_to_lds_b128(
      (glob_v4i*)(unsigned long long)gp,
      (lds_v4i*)lp,
      /*imm offset=*/0, /*cpol=*/0);
#else
  unsigned           lds_addr = (unsigned)(unsigned long long)lp;
  unsigned long long gaddr    = (unsigned long long)gp;
  asm volatile("global_load_async_to_lds_b128 %0, %1, off"
               :: "v"(lds_addr), "v"(gaddr)
               : "memory");
#endif
}

__device__ __forceinline__ void wait_asynccnt0() {
#if __has_builtin(__builtin_amdgcn_s_wait_asynccnt)
  __builtin_amdgcn_s_wait_asynccnt(0);
#else
  asm volatile("s_wait_asynccnt 0" ::: "memory");
#endif
}

// ---------------------------------------------------------------------------
__global__ __launch_bounds__(THREADS, 2)
void jacobi_fused5(const float* __restrict__ in, float* __restrict__ out) {
  extern __shared__ float smem[];
  float* cur = smem;              // TH * TX
  float* nxt = smem + TH * TX;    // TH * TX

  const int tid = threadIdx.x;
  const int tx  = tid & 31;       // lane (wave32) -> column direction
  const int ty  = tid >> 5;       // wave  -> row direction
  const int gr0 = blockIdx.y * BX;
  const int gc0 = blockIdx.x * BX;
  const int br  = gr0 - T;        // first halo row (may wrap)
  const int bcg = gc0 - 8;        // first halo col, 4-float aligned (may wrap)

  // ---- Stage 1: async b128 tile load, periodic wrap per row / col-group ----
  // A 4-aligned column group never straddles the seam (4 | 8192), so each
  // b128 transfer is contiguous in global memory even when it wraps.
  for (int idx = tid; idx < TH * CGRP; idx += THREADS) {
    const int r  = idx / CGRP;             // 0..137
    const int g  = idx - r * CGRP;         // 0..35
    const int gr = (br + r) & MASK;
    const int gc = (bcg + 4 * g) & MASK;
    async_load_b128_to_lds(in + ((long long)gr << 13) + gc, cur + r * TX + 4 * g);
  }
  wait_asynccnt0();   // my wave's async transfers have landed in LDS
  __syncthreads();    // all waves' transfers visible

  // ---- Stage 2: T fused sweeps, 2 rows per lane with packed f32 math ----
  // After sweep s the region rows [s,138-s) x cols [s,144-s) is exact; the
  // stored center (rows 5..132, cols 8..135) is exact after 5 sweeps.
  for (int t = 0; t < T; ++t) {
    for (int p = ty; p < (TH - 2) / 2; p += 8) {   // 68 row pairs
      const int    r  = 1 + 2 * p;                 // rows r, r+1
      const float* rm = cur + (r - 1) * TX;
      const float* r1 = cur + (r    ) * TX;
      const float* r2 = cur + (r + 1) * TX;
      const float* r3 = cur + (r + 2) * TX;
      float*       o1 = nxt + (r    ) * TX;
      float*       o2 = nxt + (r + 1) * TX;
      for (int c = 1 + tx; c < TX - 1; c += 32) {
        const float a0 = rm[c], a1 = r1[c], a2 = r2[c], a3 = r3[c];
        const v2f ctr = {a1, a2};                  // x = row r, y = row r+1
        const v2f up  = {a0, a1};
        const v2f dn  = {a2, a3};
        const v2f lf  = {r1[c - 1], r2[c - 1]};
        const v2f rt  = {r1[c + 1], r2[c + 1]};
        // matches reference order: x + up + down + left + right, then *0.2
        const v2f res = (ctr + up + dn + lf + rt) * 0.2f;
        o1[c] = res.x;
        o2[c] = res.y;
      }
    }
    __syncthreads();
    float* tmp = cur; cur = nxt; nxt = tmp;
  }

  // ---- Stage 3: store center 128x128 as float4 (LDS + global both 16B) ----
  for (int r = ty; r < BX; r += 8) {
    const float4 v = *(const float4*)(cur + (r + T) * TX + 8 + 4 * tx);
    *(float4*)(out + (long long)(gr0 + r) * N + gc0 + 4 * tx) = v;
  }
}

// ---------------------------------------------------------------------------
extern "C" void kernel_launch(void* const* d_in, const int* in_sizes, int n_in,
                              void* d_out, int out_size, void* d_ws, size_t ws_size,
                              hipStream_t stream) {
  (void)in_sizes; (void)n_in; (void)out_size; (void)ws_size;
  const float* x   = (const float*)d_in[0];
  float*       out = (float*)d_out;
  float*       ws  = (float*)d_ws;   // needs >= 256 MB: one ping-pong grid

  static_assert(SMEM_BYTES <= 320u * 1024u, "LDS budget exceeded");

  // Opt in to >64KB dynamic LDS (idempotent; not a stream op, capture-safe).
  (void)hipFuncSetAttribute(reinterpret_cast<const void*>(jacobi_fused5),
                            hipFuncAttributeMaxDynamicSharedMemorySize,
                            (int)SMEM_BYTES);

  dim3 grid(NTILES, NTILES), block(THREADS);
  // 20 steps = 4 launches of 5 fused steps, ping-ponging ws <-> out.
  jacobi_fused5<<<grid, block, SMEM_BYTES, stream>>>(x,   ws);
  jacobi_fused5<<<grid, block, SMEM_BYTES, stream>>>(ws,  out);
  jacobi_fused5<<<grid, block, SMEM_BYTES, stream>>>(out, ws);
  jacobi_fused5<<<grid, block, SMEM_BYTES, stream>>>(ws,  out);
}